// MLStateLinearAttnDecodeStep_10969346474370
// MI455X (gfx1250) — compile-verified
//
#include <hip/hip_runtime.h>
#include <hip/hip_bf16.h>
#include <math.h>

// ---------------------------------------------------------------------------
// MLStateLinearAttnDecodeStep for MI455X (gfx1250, wave32, WMMA + TDM)
//
// Shapes: NUM_V=32, NUM_K=16, DK=DV=128, K=4, H=2048, L=24, LI=0
// Memory-bound (~245 MB -> ~10.5us @ 23.3 TB/s):
//   - GEMVs use f32-native V_WMMA_F32_16X16X4_F32 (full precision).
//   - x vector staged to LDS once per block via TENSOR_LOAD_TO_LDS (TDM),
//     synchronized with s_wait_tensorcnt.
//   - 53 MB state pass-through copied with non-temporal hints (don't thrash
//     L2 that the weight streams want).
// ---------------------------------------------------------------------------

#define NUM_V    32
#define NUM_K    16
#define DK       128
#define DV       128
#define CONVK    4
#define H_DIM    2048
#define L_LAYERS 24
#define LI_IDX   0
#define KEY_DIM  (NUM_K * DK)               // 2048
#define VALUE_DIM (NUM_V * DV)              // 4096
#define CONV_DIM (2 * KEY_DIM + VALUE_DIM)  // 8192
#define EPS_F    1e-6f

typedef float v2f __attribute__((ext_vector_type(2)));
typedef float v4f __attribute__((ext_vector_type(4)));
typedef float v8f __attribute__((ext_vector_type(8)));
typedef unsigned u32x4 __attribute__((ext_vector_type(4)));
typedef unsigned u32x8 __attribute__((ext_vector_type(8)));

__device__ __forceinline__ float sigmoidf_(float x) { return 1.0f / (1.0f + expf(-x)); }
__device__ __forceinline__ float siluf_(float x)    { return x * sigmoidf_(x); }

// ---------------------------------------------------------------------------
// GEMV y[M] = W[M x Kd] * x[Kd] using V_WMMA_F32_16X16X4_F32.
// x is staged into LDS by a single TDM descriptor issued by wave 0.
//
// ISA VGPR layouts (CDNA5 ISA):
//   A (16x4, MxK):  lane<16 -> M=lane,    v[0]=K0, v[1]=K1
//                   lane>=16 -> M=lane-16, v[0]=K2, v[1]=K3
//   B (4x16, KxN):  same K->(vgpr, lane-half) mapping, N=lane&15
//   D (16x16):      v[r] = (M=r + 8*(lane>=16), N=lane&15)
// ---------------------------------------------------------------------------
__global__ void gemv_wmma_f32(const float* __restrict__ W,
                              const float* __restrict__ x,
                              float* __restrict__ y,
                              int M, int Kd) {
  __shared__ float xs[4096];                 // 16 KB, Kd <= 4096

  const int wave = threadIdx.x >> 5;
  const int lane = threadIdx.x & 31;

  // ---- Stage x -> LDS with the Tensor Data Mover (one DMA per block) ----
  if (wave == 0) {
    // D# group 0 (CDNA5 ISA):
    //   [1:0]=count=1, [63:32]=lds_addr, [120:64]=global_addr, [127:126]=type=2
    const unsigned long long ga = (unsigned long long)(const void*)x;
    const unsigned lds0 = (unsigned)(unsigned long long)(void*)xs;  // LDS byte offset
    u32x4 g0;
    g0[0] = 1u;
    g0[1] = lds0;
    g0[2] = (unsigned)(ga & 0xffffffffull);
    g0[3] = (unsigned)((ga >> 32) & 0x1ffffffull) | (2u << 30);
    // D# group 1 (§8.4): data_size=4B, tensor_dim0=Kd, tensor_dim1=1,
    //   tile_dim0=Kd, tile_dim1=1, tensor_dim0_stride=Kd, no pad/iterate.
    const unsigned kd = (unsigned)Kd;
    u32x8 g1;
    g1[0] = 2u << 16;                        // data_size=2 (4 bytes); wg_mask=0
    g1[1] = (kd & 0xffffu) << 16;            // tensor_dim0[15:0]
    g1[2] = ((kd >> 16) & 0xffffu) | (1u << 16); // tensor_dim0[31:16] | tensor_dim1[15:0]=1
    g1[3] = (kd & 0xffffu) << 16;            // tensor_dim1[31:16]=0 | tile_dim0=Kd
    g1[4] = 1u;                              // tile_dim1=1, tile_dim2=0
    g1[5] = kd;                              // tensor_dim0_stride[31:0]
    g1[6] = 0u;                              // stride hi / dim1_stride lo
    g1[7] = 0u;
    asm volatile("tensor_load_to_lds %0, %1" :: "s"(g0), "s"(g1) : "memory");
    __builtin_amdgcn_s_wait_tensorcnt(0);    // wave 0's TENSORcnt -> 0
  }
  __syncthreads();                           // publish LDS to all waves

  const int row0 = (blockIdx.x * (blockDim.x >> 5) + wave) * 16;
  if (row0 < M) {                            // wave-uniform: EXEC all-1s at WMMA
    const int m    = lane & 15;
    const int koff = (lane >> 4) << 1;       // 0 (lanes 0-15) or 2 (lanes 16-31)
    const float* wrow = W + (size_t)(row0 + m) * (size_t)Kd + koff;
    const float* xrow = xs + koff;           // LDS

    v8f c;
    #pragma unroll
    for (int i = 0; i < 8; ++i) c[i] = 0.0f;

    for (int k = 0; k < Kd; k += 32) {       // Kd % 32 == 0: no remainder
      #pragma unroll
      for (int u = 0; u < 8; ++u) {
        v2f a = *(const v2f*)(wrow + k + 4 * u);   // global, 8B aligned
        v2f b = *(const v2f*)(xrow + k + 4 * u);   // LDS (ds_load_b64)
        c = __builtin_amdgcn_wmma_f32_16x16x4_f32(
                false, a, false, b, (short)0, c, false, false);
      }
    }

    // Every D column is identical; lane 0 writes rows 0..7, lane 16 rows 8..15.
    if ((lane & 15) == 0) {
      const int base = row0 + ((lane >> 4) << 3);
      #pragma unroll
      for (int r = 0; r < 8; ++r) y[base + r] = c[r];
    }
  }
}

// ---------------------------------------------------------------------------
// Bulk f32 pass-through copy: non-temporal 128-bit moves (don't pollute L2).
// ---------------------------------------------------------------------------
__global__ void copy_f4(const v4f* __restrict__ src, v4f* __restrict__ dst, long n4) {
  long i = (long)blockIdx.x * blockDim.x + threadIdx.x;
  const long stride = (long)gridDim.x * blockDim.x;
  for (; i < n4; i += stride) {
    v4f t = __builtin_nontemporal_load(&src[i]);
    __builtin_nontemporal_store(t, &dst[i]);
  }
}

// ---------------------------------------------------------------------------
// Conv state shift + depthwise conv + SiLU. One thread per channel.
// ---------------------------------------------------------------------------
__global__ void conv_kernel(const float* __restrict__ conv_states,
                            const float* __restrict__ mixed_qkv,
                            const float* __restrict__ conv_w,
                            float* __restrict__ out_conv_states,
                            float* __restrict__ conv_out) {
  const int c = blockIdx.x * blockDim.x + threadIdx.x;
  if (c >= CONV_DIM) return;
  const size_t lbase = (size_t)LI_IDX * CONV_DIM * CONVK;
  const float* cs = conv_states + lbase + (size_t)c * CONVK;
  const float s0 = cs[1], s1 = cs[2], s2 = cs[3], s3 = mixed_qkv[c];
  float* oc = out_conv_states + lbase + (size_t)c * CONVK;
  oc[0] = s0; oc[1] = s1; oc[2] = s2; oc[3] = s3;
  const float* w = conv_w + (size_t)c * CONVK;
  const float acc = s0 * w[0] + s1 * w[1] + s2 * w[2] + s3 * w[3];
  conv_out[c] = siluf_(acc);
}

// ---------------------------------------------------------------------------
// Per-head scalars: beta = sigmoid(b), g_exp = exp(-exp(A_log)*softplus(a+dt))
// ---------------------------------------------------------------------------
__global__ void scalars_kernel(const float* __restrict__ b, const float* __restrict__ a,
                               const float* __restrict__ dt_bias, const float* __restrict__ A_log,
                               float* __restrict__ beta, float* __restrict__ g_exp) {
  const int h = threadIdx.x;
  if (h < NUM_V) {
    beta[h] = sigmoidf_(b[h]);
    const float xx = a[h] + dt_bias[h];
    const float sp = (xx > 20.0f) ? xx : log1pf(expf(xx));
    g_exp[h] = expf(-expf(A_log[h]) * sp);
  }
}

// ---------------------------------------------------------------------------
// L2-normalize q and k rows (head repeat V_PER_K=2). One block per head.
// ---------------------------------------------------------------------------
__global__ void qknorm_kernel(const float* __restrict__ conv_out,
                              float* __restrict__ qn, float* __restrict__ kn) {
  __shared__ float red[DK];
  const int h = blockIdx.x;
  const int v = threadIdx.x;
  const int src = (h >> 1) * DK + v;  // heads 2i,2i+1 share q/k row i
  const float qv = conv_out[src];
  const float kv = conv_out[KEY_DIM + src];

  red[v] = qv * qv; __syncthreads();
  #pragma unroll
  for (int s = 64; s > 0; s >>= 1) { if (v < s) red[v] += red[v + s]; __syncthreads(); }
  const float qs = red[0]; __syncthreads();

  red[v] = kv * kv; __syncthreads();
  #pragma unroll
  for (int s = 64; s > 0; s >>= 1) { if (v < s) red[v] += red[v + s]; __syncthreads(); }
  const float ks = red[0];

  qn[h * DK + v] = qv * rsqrtf(qs + EPS_F) * 0.08838834764831845f;  // 1/sqrt(128)
  kn[h * DK + v] = kv * rsqrtf(ks + EPS_F);
}

// ---------------------------------------------------------------------------
// Recurrent delta-rule update per head. Block = head, thread = DV column.
//   kv_mem[v] = g_exp * sum_k rec[k,v]*kn[k]
//   delta[v]  = (val[v]-kv_mem[v])*beta
//   rec'      = g_exp*rec + kn[k]*delta[v]   -> written to out (layer LI)
//   core[v]   = sum_k rec'[k,v]*qn[k]
// ---------------------------------------------------------------------------
__global__ void head_kernel(const float* __restrict__ rec_states,
                            const float* __restrict__ vvec,
                            const float* __restrict__ qn,
                            const float* __restrict__ kn,
                            const float* __restrict__ beta,
                            const float* __restrict__ g_exp,
                            float* __restrict__ out_rec_states,
                            float* __restrict__ core) {
  const int h = blockIdx.x;
  const int v = threadIdx.x;
  const float ge = g_exp[h];
  const float be = beta[h];
  const size_t base = ((size_t)LI_IDX * NUM_V + h) * (size_t)(DK * DV);
  const float* R  = rec_states + base;
  float*       Ro = out_rec_states + base;
  const float* kh = kn + h * DK;
  const float* qh = qn + h * DK;

  float kv = 0.0f;
  #pragma unroll 8
  for (int k = 0; k < DK; ++k) kv += R[(size_t)k * DV + v] * kh[k];
  kv *= ge;

  const float delta = (vvec[h * DV + v] - kv) * be;

  float acc = 0.0f;
  #pragma unroll 8
  for (int k = 0; k < DK; ++k) {
    const float r = R[(size_t)k * DV + v] * ge + kh[k] * delta;
    Ro[(size_t)k * DV + v] = r;
    acc += r * qh[k];
  }
  core[h * DV + v] = acc;
}

// ---------------------------------------------------------------------------
// RMS-norm over DV + SiLU(z) gating. One block per head.
// ---------------------------------------------------------------------------
__global__ void gate_kernel(const float* __restrict__ core,
                            const float* __restrict__ z,
                            const float* __restrict__ norm_w,
                            float* __restrict__ gated) {
  __shared__ float red[DV];
  const int h = blockIdx.x;
  const int v = threadIdx.x;
  const float cv = core[h * DV + v];
  red[v] = cv * cv; __syncthreads();
  #pragma unroll
  for (int s = 64; s > 0; s >>= 1) { if (v < s) red[v] += red[v + s]; __syncthreads(); }
  const float var = red[0] * (1.0f / DV);
  const float xn = cv * rsqrtf(var + EPS_F) * norm_w[v];
  const float zf = z[h * DV + v];
  gated[h * DV + v] = xn * siluf_(zf);
}

// ---------------------------------------------------------------------------
extern "C" void kernel_launch(void* const* d_in, const int* in_sizes, int n_in,
                              void* d_out, int out_size, void* d_ws, size_t ws_size,
                              hipStream_t stream) {
  const float* hidden_in   = (const float*)d_in[0];
  const float* conv_states = (const float*)d_in[1];   // (24,1,8192,4)
  const float* rec_states  = (const float*)d_in[2];   // (24,32,128,128)
  const float* w_qkv  = (const float*)d_in[3];        // (8192,2048)
  const float* w_z    = (const float*)d_in[4];        // (4096,2048)
  const float* w_b    = (const float*)d_in[5];        // (32,2048)
  const float* w_a    = (const float*)d_in[6];        // (32,2048)
  const float* w_out  = (const float*)d_in[7];        // (2048,4096)
  const float* conv_w = (const float*)d_in[8];        // (8192,4)
  const float* dt_bias = (const float*)d_in[9];       // (32,)
  const float* A_log   = (const float*)d_in[10];      // (32,)
  const float* norm_w  = (const float*)d_in[11];      // (128,)

  // Output: [hidden_out (2048) | new_conv_states (786432) | new_rec_states (12582912)]
  float* out = (float*)d_out;
  float* hid_out        = out;
  float* conv_out_state = out + 2048;
  float* rec_out_state  = out + 2048 + (long)L_LAYERS * CONV_DIM * CONVK;

  // Workspace layout (floats)
  float* ws    = (float*)d_ws;
  float* qkv   = ws;            // 8192
  float* zv    = ws + 8192;     // 4096
  float* bv    = ws + 12288;    // 32
  float* av    = ws + 12320;    // 32
  float* convo = ws + 12352;    // 8192  (conv_out: q|k|v)
  float* qn    = ws + 20544;    // 4096
  float* kn    = ws + 24640;    // 4096
  float* beta  = ws + 28736;    // 32
  float* gexp  = ws + 28768;    // 32
  float* core  = ws + 28800;    // 4096
  float* gated = ws + 32896;    // 4096

  const long conv_n4 = (long)L_LAYERS * CONV_DIM * CONVK / 4;     // 196608
  const long rec_n4  = (long)L_LAYERS * NUM_V * DK * DV / 4;      // 3145728

  // 1) State pass-through copies (layer LI gets overwritten later in-stream).
  copy_f4<<<1024, 256, 0, stream>>>((const v4f*)conv_states, (v4f*)conv_out_state, conv_n4);
  copy_f4<<<2048, 256, 0, stream>>>((const v4f*)rec_states,  (v4f*)rec_out_state,  rec_n4);

  // 2) Projections (WMMA f32 GEMVs, TDM-staged x). 8 waves * 16 rows / block.
  gemv_wmma_f32<<<64, 256, 0, stream>>>(w_qkv, hidden_in, qkv, CONV_DIM, H_DIM);
  gemv_wmma_f32<<<32, 256, 0, stream>>>(w_z,   hidden_in, zv,  VALUE_DIM, H_DIM);
  gemv_wmma_f32<<<1,  256, 0, stream>>>(w_b,   hidden_in, bv,  NUM_V, H_DIM);
  gemv_wmma_f32<<<1,  256, 0, stream>>>(w_a,   hidden_in, av,  NUM_V, H_DIM);

  // 3) Conv shift + SiLU; writes new conv state at layer LI.
  conv_kernel<<<CONV_DIM / 256, 256, 0, stream>>>(conv_states, qkv, conv_w, conv_out_state, convo);

  // 4) Gate scalars, q/k normalization.
  scalars_kernel<<<1, 32, 0, stream>>>(bv, av, dt_bias, A_log, beta, gexp);
  qknorm_kernel<<<NUM_V, DK, 0, stream>>>(convo, qn, kn);

  // 5) Recurrent delta-rule update per head; writes new rec state at layer LI.
  head_kernel<<<NUM_V, DV, 0, stream>>>(rec_states, convo + 2 * KEY_DIM, qn, kn, beta, gexp,
                                        rec_out_state, core);

  // 6) RMS-norm + SiLU(z) gate, then output projection (WMMA f32 GEMV).
  gate_kernel<<<NUM_V, DV, 0, stream>>>(core, zv, norm_w, gated);
  gemv_wmma_f32<<<16, 256, 0, stream>>>(w_out, gated, hid_out, H_DIM, VALUE_DIM);
}